// SparseMMGATLayer_21741124452467
// MI455X (gfx1250) — compile-verified
//
#include <hip/hip_runtime.h>

#define NEG_SLOPE 0.2f
#define LN_EPS 1e-5f
#define D 64

typedef __attribute__((ext_vector_type(2))) float v2f;
typedef __attribute__((ext_vector_type(8))) float v8f;

// ---- order-preserving float <-> uint encoding for atomic max ----
__device__ __forceinline__ unsigned enc_f32(float f) {
  int i = __float_as_int(f);
  return (i >= 0) ? ((unsigned)i | 0x80000000u) : ~(unsigned)i;
}
__device__ __forceinline__ float dec_f32(unsigned k) {
  int i = (k & 0x80000000u) ? (int)(k & 0x7FFFFFFFu) : (int)~k;
  return __int_as_float(i);
}

// ---------------------------------------------------------------------------
// K0: init workspace (h_prime = 0, denom = 0, m_enc = enc(-inf))
// ---------------------------------------------------------------------------
__global__ void k_init(float* __restrict__ hprime, float* __restrict__ denom,
                       unsigned* __restrict__ menc, int N) {
  int i = blockIdx.x * blockDim.x + threadIdx.x;
  if (i < N * D) hprime[i] = 0.0f;
  if (i < N) {
    denom[i] = 0.0f;
    menc[i]  = 0x007FFFFFu;   // enc(-inf)
  }
}

// ---------------------------------------------------------------------------
// K1: hW = h @ W^T via V_WMMA_F32_16X16X4_F32.
// One wave (32 lanes) per 16-row tile; 4 column tiles of 16; 16 K-steps of 4.
// A-frag (16x4 f32): lanes 0-15 -> M=lane, VGPR0=K0,VGPR1=K1; lanes 16-31 K2/K3.
// B-frag (4x16 f32): lanes 0-15 -> N=lane, VGPR0=K0,VGPR1=K1; lanes 16-31 K2/K3.
// C/D (16x16 f32): lane half selects M base (0 or 8), VGPR v = row offset.
// ---------------------------------------------------------------------------
__global__ void k_gemm(const float* __restrict__ h, const float* __restrict__ W,
                       float* __restrict__ hW) {
  const int row0 = blockIdx.x * 16;
  const int lane = threadIdx.x;          // 0..31, blockDim.x == 32
  const int m    = lane & 15;
  const int kh   = lane >> 4;            // 0 or 1

  v8f acc[4] = {{}, {}, {}, {}};
  const float* hrow = h + (size_t)(row0 + m) * D;

#pragma unroll
  for (int k0 = 0; k0 < D; k0 += 4) {
    const int ka = k0 + kh * 2;
    v2f A;
    A.x = hrow[ka];
    A.y = hrow[ka + 1];
#pragma unroll
    for (int c = 0; c < 4; ++c) {
      // B = W^T  =>  B[k][n] = W[n][k],  n = c*16 + m
      const float* wrow = W + (size_t)(c * 16 + m) * D;
      v2f B;
      B.x = wrow[ka];
      B.y = wrow[ka + 1];
      acc[c] = __builtin_amdgcn_wmma_f32_16x16x4_f32(
          /*neg_a=*/false, A, /*neg_b=*/false, B,
          /*c_mod=*/(short)0, acc[c], /*reuse_a=*/false, /*reuse_b=*/false);
    }
  }

#pragma unroll
  for (int c = 0; c < 4; ++c)
#pragma unroll
    for (int v = 0; v < 8; ++v)
      hW[(size_t)(row0 + kh * 8 + v) * D + c * 16 + m] = acc[c][v];
}

// ---------------------------------------------------------------------------
// K2: s1[n] = hW[n,:] . a[0:64],  s2[n] = hW[n,:] . a[64:128]
// One wave per node, 2 elements per lane, butterfly reduce.
// ---------------------------------------------------------------------------
__global__ void k_scores(const float* __restrict__ hW, const float* __restrict__ a,
                         float* __restrict__ s1, float* __restrict__ s2, int N) {
  int node = blockIdx.x * (blockDim.x >> 5) + (threadIdx.x >> 5);
  if (node >= N) return;
  int lane = threadIdx.x & 31;
  size_t base = (size_t)node * D;
  float x0 = hW[base + lane];
  float x1 = hW[base + lane + 32];
  float p1 = x0 * a[lane]      + x1 * a[lane + 32];
  float p2 = x0 * a[64 + lane] + x1 * a[96 + lane];
#pragma unroll
  for (int off = 16; off; off >>= 1) {
    p1 += __shfl_xor(p1, off, 32);
    p2 += __shfl_xor(p2, off, 32);
  }
  if (lane == 0) {
    s1[node] = p1;
    s2[node] = p2;
  }
}

// ---------------------------------------------------------------------------
// K3: per-edge  e = leaky_relu(s1[src] + s2[dst]); segment max over src
// ---------------------------------------------------------------------------
__global__ void k_edge1(const int* __restrict__ ei, const float* __restrict__ s1,
                        const float* __restrict__ s2, float* __restrict__ e,
                        unsigned* __restrict__ menc, int E) {
  int i = blockIdx.x * blockDim.x + threadIdx.x;
  if (i >= E) return;
  int src = ei[i];
  int dst = ei[E + i];
  float v = s1[src] + s2[dst];
  v = (v > 0.0f) ? v : NEG_SLOPE * v;
  e[i] = v;
  atomicMax(&menc[src], enc_f32(v));
}

// ---------------------------------------------------------------------------
// K4: ex = exp(e - m[src]);  denom[src] += ex   (ex overwrites e in place)
// ---------------------------------------------------------------------------
__global__ void k_edge2(const int* __restrict__ ei, const unsigned* __restrict__ menc,
                        float* __restrict__ e, float* __restrict__ denom, int E) {
  int i = blockIdx.x * blockDim.x + threadIdx.x;
  if (i >= E) return;
  int src = ei[i];
  float mv = dec_f32(menc[src]);
  float ex = __expf(e[i] - mv);
  e[i] = ex;
  atomicAdd(&denom[src], ex);
}

// ---------------------------------------------------------------------------
// K5: scatter  h_prime[dst] += (ex/denom[src]) * hW[src]
// 64 consecutive lanes handle the 64 dims of one edge (coalesced).
// ---------------------------------------------------------------------------
__global__ void k_scatter(const int* __restrict__ ei, const float* __restrict__ e,
                          const float* __restrict__ denom, const float* __restrict__ hW,
                          float* __restrict__ hprime, int E) {
  int idx = blockIdx.x * blockDim.x + threadIdx.x;
  int i = idx >> 6;
  if (i >= E) return;
  int d = idx & 63;
  int src = ei[i];
  int dst = ei[E + i];
  float att = e[i] / denom[src];
  atomicAdd(&hprime[(size_t)dst * D + d], att * hW[(size_t)src * D + d]);
}

// ---------------------------------------------------------------------------
// K6: x = hW + h_prime; LayerNorm over last dim; write d_out
// One wave per node, 2 elements per lane.
// ---------------------------------------------------------------------------
__global__ void k_ln(const float* __restrict__ hW, const float* __restrict__ hprime,
                     const float* __restrict__ gamma, const float* __restrict__ beta,
                     float* __restrict__ out, int N) {
  int node = blockIdx.x * (blockDim.x >> 5) + (threadIdx.x >> 5);
  if (node >= N) return;
  int lane = threadIdx.x & 31;
  size_t base = (size_t)node * D;
  float x0 = hW[base + lane]      + hprime[base + lane];
  float x1 = hW[base + lane + 32] + hprime[base + lane + 32];
  float s  = x0 + x1;
  float ss = x0 * x0 + x1 * x1;
#pragma unroll
  for (int off = 16; off; off >>= 1) {
    s  += __shfl_xor(s, off, 32);
    ss += __shfl_xor(ss, off, 32);
  }
  float mu  = s * (1.0f / 64.0f);
  float var = ss * (1.0f / 64.0f) - mu * mu;
  float r   = rsqrtf(var + LN_EPS);
  out[base + lane]      = (x0 - mu) * r * gamma[lane]      + beta[lane];
  out[base + lane + 32] = (x1 - mu) * r * gamma[lane + 32] + beta[lane + 32];
}

// ---------------------------------------------------------------------------
extern "C" void kernel_launch(void* const* d_in, const int* in_sizes, int n_in,
                              void* d_out, int out_size, void* d_ws, size_t ws_size,
                              hipStream_t stream) {
  const float* h     = (const float*)d_in[0];
  const int*   ei    = (const int*)d_in[1];
  const float* W     = (const float*)d_in[2];
  const float* a     = (const float*)d_in[3];
  const float* gamma = (const float*)d_in[4];
  const float* beta  = (const float*)d_in[5];
  float* out = (float*)d_out;

  const int N = in_sizes[0] / D;   // 50000
  const int E = in_sizes[1] / 2;   // 1000000

  // workspace layout
  char* ws = (char*)d_ws;
  float* hW = (float*)ws;         ws += (size_t)N * D * sizeof(float);
  float* hprime = (float*)ws;     ws += (size_t)N * D * sizeof(float);
  float* s1 = (float*)ws;         ws += (size_t)N * sizeof(float);
  float* s2 = (float*)ws;         ws += (size_t)N * sizeof(float);
  float* denom = (float*)ws;      ws += (size_t)N * sizeof(float);
  unsigned* menc = (unsigned*)ws; ws += (size_t)N * sizeof(unsigned);
  float* e = (float*)ws;          ws += (size_t)E * sizeof(float);

  const int BT = 256;

  // init (independent of gemm, both precede the edge stage; stream order serializes)
  k_init<<<(N * D + BT - 1) / BT, BT, 0, stream>>>(hprime, denom, menc, N);

  // dense GEMM with WMMA: one wave per 16-row tile (N % 16 == 0 here)
  k_gemm<<<N / 16, 32, 0, stream>>>(h, W, hW);

  // attention scores: one wave per node (8 waves / 256-thread block)
  k_scores<<<(N + 7) / 8, BT, 0, stream>>>(hW, a, s1, s2, N);

  // per-edge softmax (segment max, then exp + segment sum)
  k_edge1<<<(E + BT - 1) / BT, BT, 0, stream>>>(ei, s1, s2, e, menc, E);
  k_edge2<<<(E + BT - 1) / BT, BT, 0, stream>>>(ei, menc, e, denom, E);

  // message scatter: 64 lanes per edge
  long long sc_threads = (long long)E * D;
  k_scatter<<<(unsigned)((sc_threads + BT - 1) / BT), BT, 0, stream>>>(
      ei, e, denom, hW, hprime, E);

  // residual + LayerNorm
  k_ln<<<(N + 7) / 8, BT, 0, stream>>>(hW, hprime, gamma, beta, out, N);
}